// AsymmetricKVBudgetReadout_77077483094336
// MI455X (gfx1250) — compile-verified
//
#include <hip/hip_runtime.h>
#include <hip/hip_bf16.h>
#include <math.h>

// ---------------------------------------------------------------------------
// AsymmetricKVBudgetReadout for MI455X (gfx1250, wave32, WMMA + async-to-LDS)
//
// Shapes: N=512, S=2048, DIM=256, CLASSES=1000, route k=32, value k=128
// Pipeline:
//   1) wmma_gemm_atb:  qp = q @ Wq^T                        (WMMA f32 16x16x4)
//   2) logits_kernel:  route/value logits, ONE pass over K and V.
//        K/V rows staged in LDS via global_load_async_to_lds_b128 (ASYNCcnt),
//        WMMA B fragments read from LDS with a 4-float row pad -> conflict-free
//   3) topk_kernel:    exact top-k via 32-step radix search on float bits;
//        masked softmax; dense weights output + compact (idx,w) lists
//   4) summary_kernel: gather-weighted sums of V rows (+prefetch), gate
//   5) wmma_gemm_atb:  logits = summary @ Wc^T + bc ; recon = summary@Wr^T+br
// ---------------------------------------------------------------------------

typedef float v2f __attribute__((ext_vector_type(2)));
typedef float v8f __attribute__((ext_vector_type(8)));

#define NQ 512
#define SS 2048
#define DD 256
#define NCLS 1000
#define KR 32
#define KV 128
#define ROWF 260  // 256 floats + 4 pad: ds_load_b64 fragment reads hit 64 distinct banks

// async global -> LDS, 16 bytes per lane (512 B per wave op), tracked by ASYNCcnt
__device__ __forceinline__ void async_ld_b128(unsigned lds_off, const void* gaddr) {
  asm volatile("global_load_async_to_lds_b128 %0, %1, off"
               :: "v"(lds_off), "v"((unsigned long long)(uintptr_t)gaddr)
               : "memory");
}
__device__ __forceinline__ void wait_async0() {
  asm volatile("s_wait_asynccnt 0x0" ::: "memory");
}

// ---------------------------------------------------------------------------
// GEMM: out[m][j] = sum_k A[m][k] * W[j][k] + bias[j],  K fixed = 256.
// One wave32 per 16x16 output tile; 64 WMMA_F32_16X16X4_F32 per tile.
// A frag (16x4): lanes 0-15 rows M=0..15 K={k0,k0+1}; lanes 16-31 K={k0+2,k0+3}.
// C/D (16x16):   VGPR r -> row m0+r (lanes 0-15) / m0+8+r (lanes 16-31).
// ---------------------------------------------------------------------------
__global__ void wmma_gemm_atb(const float* __restrict__ A,
                              const float* __restrict__ W,
                              const float* __restrict__ bias,
                              float* __restrict__ out,
                              int M, int N) {
  const int lane = threadIdx.x;          // one full wave32
  const int half = lane >> 4;
  const int l15  = lane & 15;
  const int n0 = blockIdx.x * 16;
  const int m0 = blockIdx.y * 16;

  int jc = n0 + l15; if (jc >= N) jc = N - 1;   // clamp B column (masked at store)
  const float* aRow = A + (size_t)(m0 + l15) * DD + half * 2;
  const float* wRow = W + (size_t)jc * DD + half * 2;

  v8f acc = {};
#pragma unroll 4
  for (int k0 = 0; k0 < DD; k0 += 4) {
    v2f a = *(const v2f*)(aRow + k0);
    v2f b = *(const v2f*)(wRow + k0);
    acc = __builtin_amdgcn_wmma_f32_16x16x4_f32(false, a, false, b,
                                                (short)0, acc, false, false);
  }

  const int col = n0 + l15;
  if (col < N) {
    const float bv = bias ? bias[col] : 0.0f;
#pragma unroll
    for (int r = 0; r < 8; ++r) {
      const int row = m0 + r + half * 8;
      out[(size_t)row * N + col] = acc[r] + bv;
    }
  }
}

// ---------------------------------------------------------------------------
// Logits pass: one wave32 per (n, 16-wide s block).
//   - 16 K rows + 16 V rows (+ qp[n]) async-copied to LDS, fully coalesced
//   - A = qp[n] broadcast to all 16 rows (LDS broadcast reads)
//   - B = 16 staged rows; padded stride makes ds_load_b64 conflict-free
//   - every D row equals the logit vector; row 0 (lanes 0-15) is stored
// ---------------------------------------------------------------------------
__global__ void logits_kernel(const float* __restrict__ Kg,
                              const float* __restrict__ Vg,
                              const float* __restrict__ qp,
                              float* __restrict__ rl,
                              float* __restrict__ vl) {
  __shared__ __align__(16) float ldsK[16 * ROWF];
  __shared__ __align__(16) float ldsV[16 * ROWF];
  __shared__ __align__(16) float ldsQ[DD];

  const int lane = threadIdx.x;
  const int half = lane >> 4;
  const int l15  = lane & 15;
  const int s0 = blockIdx.x * 16;
  const int n  = blockIdx.y;

  const char* gK = (const char*)(Kg + ((size_t)n * SS + s0) * DD);
  const char* gV = (const char*)(Vg + ((size_t)n * SS + s0) * DD);
  const char* gQ = (const char*)(qp + (size_t)n * DD);
  const unsigned lK = (unsigned)(uintptr_t)&ldsK[0];
  const unsigned lV = (unsigned)(uintptr_t)&ldsV[0];
  const unsigned lQ = (unsigned)(uintptr_t)&ldsQ[0];

  // qp[n]: 1 KB in two wave-wide async ops
  async_ld_b128(lQ + lane * 16, gQ + lane * 16);
  async_ld_b128(lQ + lane * 16 + 512, gQ + lane * 16 + 512);
  // 16 rows x 1 KB for K and V (1040 B padded LDS rows)
#pragma unroll
  for (int r = 0; r < 16; ++r) {
    const unsigned lo = (unsigned)r * (ROWF * 4) + lane * 16;
    const unsigned go = (unsigned)r * (DD * 4) + lane * 16;
    async_ld_b128(lK + lo,       gK + go);
    async_ld_b128(lK + lo + 512, gK + go + 512);
    async_ld_b128(lV + lo,       gV + go);
    async_ld_b128(lV + lo + 512, gV + go + 512);
  }
  wait_async0();  // single wave owns this LDS tile; no barrier needed

  const unsigned fb = (unsigned)l15 * ROWF + half * 2;  // B fragment base
  const unsigned qb = half * 2;                         // A fragment base (broadcast)

  v8f accK = {}, accV = {};
#pragma unroll 4
  for (int k0 = 0; k0 < DD; k0 += 4) {
    v2f a  = *(const v2f*)&ldsQ[qb + k0];
    v2f bk = *(const v2f*)&ldsK[fb + k0];
    v2f bv = *(const v2f*)&ldsV[fb + k0];
    accK = __builtin_amdgcn_wmma_f32_16x16x4_f32(false, a, false, bk,
                                                 (short)0, accK, false, false);
    accV = __builtin_amdgcn_wmma_f32_16x16x4_f32(false, a, false, bv,
                                                 (short)0, accV, false, false);
  }

  if (lane < 16) {                     // D row M=0 = logits for the 16 s's
    const size_t o = (size_t)n * SS + s0 + lane;
    rl[o] = accK[0] * 0.0625f;         // 1/sqrt(256)
    vl[o] = accV[0] * 0.0625f;
  }
}

// order-preserving float -> uint mapping
__device__ __forceinline__ unsigned f2u(float f) {
  unsigned u = __float_as_uint(f);
  return (u & 0x80000000u) ? ~u : (u | 0x80000000u);
}

// ---------------------------------------------------------------------------
// Per-n top-k + masked softmax + dense weights output + compact kept lists.
// One 256-thread block (8 waves) per n. Logits staged in LDS (16 KB).
// ---------------------------------------------------------------------------
__global__ void topk_kernel(const float* __restrict__ rl_g,
                            const float* __restrict__ vl_g,
                            float* __restrict__ w_out,
                            int* __restrict__ ridx, float* __restrict__ rwt,
                            int* __restrict__ vidx, float* __restrict__ vwt) {
  __shared__ float rl[SS];
  __shared__ float vl[SS];
  __shared__ float red[256];
  __shared__ unsigned cnt[2];

  const int tid = threadIdx.x;
  const int n   = blockIdx.x;
  const size_t base = (size_t)n * SS;

  for (int i = tid; i < SS; i += 256) { rl[i] = rl_g[base + i]; vl[i] = vl_g[base + i]; }
  __syncthreads();

  // Radix binary search for the k-th largest value (exact for distinct floats):
  // threshold T = max u such that count(x >= u) >= k.
  unsigned tr = 0u, tv = 0u;
  for (int bit = 31; bit >= 0; --bit) {
    if (tid == 0) { cnt[0] = 0u; cnt[1] = 0u; }
    __syncthreads();
    const unsigned cr = tr | (1u << bit), cv = tv | (1u << bit);
    unsigned lr = 0u, lv = 0u;
    for (int i = tid; i < SS; i += 256) {
      lr += (f2u(rl[i]) >= cr);
      lv += (f2u(vl[i]) >= cv);
    }
    if (lr) atomicAdd(&cnt[0], lr);
    if (lv) atomicAdd(&cnt[1], lv);
    __syncthreads();
    if (cnt[0] >= (unsigned)KR) tr = cr;
    if (cnt[1] >= (unsigned)KV) tv = cv;
    __syncthreads();
  }

  // maxes (the global max is always inside the top-k set)
  float m = -3.402823466e38f;
  for (int i = tid; i < SS; i += 256) m = fmaxf(m, rl[i]);
  red[tid] = m; __syncthreads();
  for (int off = 128; off > 0; off >>= 1) { if (tid < off) red[tid] = fmaxf(red[tid], red[tid + off]); __syncthreads(); }
  const float rmax = red[0]; __syncthreads();

  m = -3.402823466e38f;
  for (int i = tid; i < SS; i += 256) m = fmaxf(m, vl[i]);
  red[tid] = m; __syncthreads();
  for (int off = 128; off > 0; off >>= 1) { if (tid < off) red[tid] = fmaxf(red[tid], red[tid + off]); __syncthreads(); }
  const float vmax = red[0]; __syncthreads();

  // softmax denominators over kept sets
  float sr = 0.f, sv = 0.f;
  for (int i = tid; i < SS; i += 256) {
    if (f2u(rl[i]) >= tr) sr += __expf(rl[i] - rmax);
    if (f2u(vl[i]) >= tv) sv += __expf(vl[i] - vmax);
  }
  red[tid] = sr; __syncthreads();
  for (int off = 128; off > 0; off >>= 1) { if (tid < off) red[tid] += red[tid + off]; __syncthreads(); }
  const float rinv = 1.0f / red[0]; __syncthreads();
  red[tid] = sv; __syncthreads();
  for (int off = 128; off > 0; off >>= 1) { if (tid < off) red[tid] += red[tid + off]; __syncthreads(); }
  const float vinv = 1.0f / red[0]; __syncthreads();

  // dense weights output + compact kept lists
  if (tid == 0) { cnt[0] = 0u; cnt[1] = 0u; }
  __syncthreads();
  for (int i = tid; i < SS; i += 256) {
    float rw = 0.f, vw = 0.f;
    if (f2u(rl[i]) >= tr) {
      rw = __expf(rl[i] - rmax) * rinv;
      unsigned p = atomicAdd(&cnt[0], 1u);
      if (p < (unsigned)KR) { ridx[n * KR + p] = i; rwt[n * KR + p] = rw; }
    }
    if (f2u(vl[i]) >= tv) {
      vw = __expf(vl[i] - vmax) * vinv;
      unsigned p = atomicAdd(&cnt[1], 1u);
      if (p < (unsigned)KV) { vidx[n * KV + p] = i; vwt[n * KV + p] = vw; }
    }
    w_out[base + i] = 0.5f * (rw + vw);
  }
}

// ---------------------------------------------------------------------------
// Gather-weighted summaries (32 + 128 V rows per n), gate, blended summary.
// One 256-thread block per n; thread tid owns feature dim d = tid.
// global_prefetch_b8 issued a few rows ahead for the irregular gathers.
// ---------------------------------------------------------------------------
__global__ void summary_kernel(const float* __restrict__ Vg,
                               const float* __restrict__ qp,
                               const int* __restrict__ ridx, const float* __restrict__ rwt,
                               const int* __restrict__ vidx, const float* __restrict__ vwt,
                               float* __restrict__ summ) {
  __shared__ int   si[KV];
  __shared__ float sw[KV];
  __shared__ float red[256];
  const int tid = threadIdx.x;
  const int n   = blockIdx.x;
  const size_t vbase = (size_t)n * SS * DD;

  float rs = 0.f, vs = 0.f;
  if (tid < KR) { si[tid] = ridx[n * KR + tid]; sw[tid] = rwt[n * KR + tid]; }
  __syncthreads();
#pragma unroll 4
  for (int j = 0; j < KR; ++j) {
    if (j + 4 < KR) __builtin_prefetch(&Vg[vbase + (size_t)si[j + 4] * DD + tid], 0, 0);
    rs += sw[j] * Vg[vbase + (size_t)si[j] * DD + tid];
  }
  __syncthreads();
  if (tid < KV) { si[tid] = vidx[n * KV + tid]; sw[tid] = vwt[n * KV + tid]; }
  __syncthreads();
#pragma unroll 4
  for (int j = 0; j < KV; ++j) {
    if (j + 4 < KV) __builtin_prefetch(&Vg[vbase + (size_t)si[j + 4] * DD + tid], 0, 0);
    vs += sw[j] * Vg[vbase + (size_t)si[j] * DD + tid];
  }

  const float qv = qp[(size_t)n * DD + tid];
  red[tid] = (rs - vs) * qv;
  __syncthreads();
  for (int off = 128; off > 0; off >>= 1) { if (tid < off) red[tid] += red[tid + off]; __syncthreads(); }
  const float gate = 1.0f / (1.0f + __expf(-red[0] * 0.0625f));
  summ[(size_t)n * DD + tid] = gate * rs + (1.0f - gate) * vs;
}

// ---------------------------------------------------------------------------
extern "C" void kernel_launch(void* const* d_in, const int* in_sizes, int n_in,
                              void* d_out, int out_size, void* d_ws, size_t ws_size,
                              hipStream_t stream) {
  const float* q  = (const float*)d_in[0];
  const float* Kg = (const float*)d_in[1];
  const float* Vg = (const float*)d_in[2];
  const float* Wq = (const float*)d_in[3];
  const float* Wc = (const float*)d_in[4];
  const float* bc = (const float*)d_in[5];
  const float* Wr = (const float*)d_in[6];
  const float* br = (const float*)d_in[7];

  float* out        = (float*)d_out;
  float* out_logits = out;                         // [512, 1000]
  float* out_recon  = out + (size_t)NQ * NCLS;     // [512, 256]
  float* out_w      = out_recon + (size_t)NQ * DD; // [512, 2048]

  // workspace carve-up (~9.7 MB)
  float* qp   = (float*)d_ws;                      // 512*256
  float* rl   = qp + (size_t)NQ * DD;              // 512*2048
  float* vl   = rl + (size_t)NQ * SS;              // 512*2048
  int*   ridx = (int*)(vl + (size_t)NQ * SS);      // 512*32
  float* rwt  = (float*)(ridx + NQ * KR);          // 512*32
  int*   vidx = (int*)(rwt + NQ * KR);             // 512*128
  float* vwt  = (float*)(vidx + NQ * KV);          // 512*128
  float* summ = vwt + NQ * KV;                     // 512*256

  const dim3 wave(32);

  // 1) qp = q @ Wq^T
  wmma_gemm_atb<<<dim3(DD / 16, NQ / 16), wave, 0, stream>>>(q, Wq, nullptr, qp, NQ, DD);
  // 2) route/value logits (single async-staged pass over K and V)
  logits_kernel<<<dim3(SS / 16, NQ), wave, 0, stream>>>(Kg, Vg, qp, rl, vl);
  // 3) top-k thresholds, softmax weights, dense weights output
  topk_kernel<<<NQ, 256, 0, stream>>>(rl, vl, out_w, ridx, rwt, vidx, vwt);
  // 4) gathered summaries + gate
  summary_kernel<<<NQ, 256, 0, stream>>>(Vg, qp, ridx, rwt, vidx, vwt, summ);
  // 5) output heads
  wmma_gemm_atb<<<dim3((NCLS + 15) / 16, NQ / 16), wave, 0, stream>>>(summ, Wc, bc, out_logits, NQ, NCLS);
  wmma_gemm_atb<<<dim3(DD / 16, NQ / 16), wave, 0, stream>>>(summ, Wr, br, out_recon, NQ, DD);
}